// MultiBranchFusion_18949395710068
// MI455X (gfx1250) — compile-verified
//
#include <hip/hip_runtime.h>

// Problem constants (from reference)
#define NAPP   1024
#define NSP    512
#define NHID   512
#define NCARD  16
#define NSUB   32
#define NCOLS  512          // NCARD * NSUB
#define M_TILE 64           // rows per block
#define BLOCK_THREADS 128   // 4 waves (wave32), 16 rows per wave

typedef float v2f __attribute__((ext_vector_type(2)));
typedef float v8f __attribute__((ext_vector_type(8)));

__global__ __launch_bounds__(BLOCK_THREADS)
void mbf_fused_kernel(const float* __restrict__ A,   // [N, NAPP]
                      const float* __restrict__ S,   // [N, NSP]
                      const float* __restrict__ W1,  // [NCARD, NAPP, NSUB]
                      const float* __restrict__ b1,  // [NCARD, NSUB]  (flat == 512 cols)
                      const float* __restrict__ W2,  // [NCARD, NSP, NSUB]
                      const float* __restrict__ b2,  // [NCARD, NSUB]
                      const float* __restrict__ W3,  // [NCARD, NSUB, NHID] == row-major [512, 512]
                      const float* __restrict__ b3,  // [NCARD, NHID]
                      float* __restrict__ out)       // [N, NHID]
{
    extern __shared__ float smem[];
    float* hS   = smem;                         // [M_TILE][NCOLS]  fused intermediate
    float* bvec = smem + M_TILE * NCOLS;        // [NCOLS]  b1+b2
    float* b3s  = bvec + NCOLS;                 // [NHID]   sum_c b3[c]

    const int tid  = threadIdx.x;
    const int wv   = tid >> 5;          // wave id 0..3
    const int lane = tid & 31;
    const int m    = lane & 15;         // row-in-tile / col-in-tile lane mapping
    const int klo  = (lane >> 4) << 1;  // 0 for lanes 0-15, 2 for lanes 16-31

    // Phase 0: fold biases (uniform loop bounds -> no divergence)
    for (int c = tid; c < NCOLS; c += BLOCK_THREADS) {
        bvec[c] = b1[c] + b2[c];
        float s = 0.f;
        #pragma unroll
        for (int cc = 0; cc < NCARD; ++cc) s += b3[cc * NHID + c];
        b3s[c] = s;
    }
    __syncthreads();

    const int rowBase = blockIdx.x * M_TILE + wv * 16;   // first row of this wave's strip
    const float* Arow = A + (size_t)(rowBase + m) * NAPP;
    const float* Srow = S + (size_t)(rowBase + m) * NSP;

    // ---------------- Phase 1: h = relu(A@W1r + S@W2r + bvec) ----------------
    // Each j-pass covers one full branch c == j (32 columns = 2 WMMA N-tiles),
    // sharing one A fragment across both tiles: 1 b64 + 4 b32 loads per 2 wmma.
    for (int j = 0; j < NCARD; ++j) {
        // Tile pair columns: col0 = j*32 + m, col1 = col0 + 16 (same branch j)
        const float* W1c = W1 + (size_t)j * NAPP * NSUB + m;   // + k*NSUB (+16 for tile 1)
        const float* W2c = W2 + (size_t)j * NSP  * NSUB + m;

        v8f acc0 = {};
        v8f acc1 = {};
        #pragma unroll 4
        for (int kb = 0; kb < NAPP; kb += 4) {
            if ((kb & 127) == 0)   // uniform: stream-ahead prefetch of the A row
                __builtin_prefetch(Arow + kb + 256, 0, 0);
            v2f a = *(const v2f*)(Arow + kb + klo);            // K=klo, klo+1
            const float* Bb = W1c + (size_t)(kb + klo) * NSUB; // one base, 4 imm offsets
            v2f b0, b1v;
            b0.x  = Bb[0];          // col0, K=klo
            b1v.x = Bb[16];         // col1, K=klo
            b0.y  = Bb[NSUB];       // col0, K=klo+1
            b1v.y = Bb[NSUB + 16];  // col1, K=klo+1
            acc0 = __builtin_amdgcn_wmma_f32_16x16x4_f32(false, a, false, b0,
                                                         (short)0, acc0, false, false);
            acc1 = __builtin_amdgcn_wmma_f32_16x16x4_f32(false, a, false, b1v,
                                                         (short)0, acc1, false, false);
        }
        #pragma unroll 4
        for (int kb = 0; kb < NSP; kb += 4) {
            v2f a = *(const v2f*)(Srow + kb + klo);
            const float* Bb = W2c + (size_t)(kb + klo) * NSUB;
            v2f b0, b1v;
            b0.x  = Bb[0];
            b1v.x = Bb[16];
            b0.y  = Bb[NSUB];
            b1v.y = Bb[NSUB + 16];
            acc0 = __builtin_amdgcn_wmma_f32_16x16x4_f32(false, a, false, b0,
                                                         (short)0, acc0, false, false);
            acc1 = __builtin_amdgcn_wmma_f32_16x16x4_f32(false, a, false, b1v,
                                                         (short)0, acc1, false, false);
        }

        const int col0 = j * NSUB + m;
        const float bias0 = bvec[col0];
        const float bias1 = bvec[col0 + 16];
        #pragma unroll
        for (int i = 0; i < 8; ++i) {
            // C/D layout: VGPR i holds M=i (lanes 0-15) and M=i+8 (lanes 16-31)
            const int r = wv * 16 + ((lane < 16) ? i : i + 8);
            float v0 = acc0[i] + bias0;
            float v1 = acc1[i] + bias1;
            hS[r * NCOLS + col0]      = v0 > 0.f ? v0 : 0.f;
            hS[r * NCOLS + col0 + 16] = v1 > 0.f ? v1 : 0.f;
        }
    }
    __syncthreads();

    // ---------------- Phase 2: out = relu(h @ W3r + b3sum) ----------------
    const float* hRow = hS + (wv * 16 + m) * NCOLS;
    for (int j = 0; j < NHID / 32; ++j) {
        const int col0 = j * 32 + m;
        v8f acc0 = {};
        v8f acc1 = {};
        #pragma unroll 4
        for (int kb = 0; kb < NCOLS; kb += 4) {
            v2f a = *(const v2f*)(hRow + kb + klo);             // ds_load_b64
            const float* Bb = W3 + (size_t)(kb + klo) * NHID + col0;
            v2f b0, b1v;
            b0.x  = Bb[0];          // col0,    K=klo
            b1v.x = Bb[16];         // col0+16, K=klo
            b0.y  = Bb[NHID];       // col0,    K=klo+1
            b1v.y = Bb[NHID + 16];  // col0+16, K=klo+1
            acc0 = __builtin_amdgcn_wmma_f32_16x16x4_f32(false, a, false, b0,
                                                         (short)0, acc0, false, false);
            acc1 = __builtin_amdgcn_wmma_f32_16x16x4_f32(false, a, false, b1v,
                                                         (short)0, acc1, false, false);
        }
        const float bias0 = b3s[col0];
        const float bias1 = b3s[col0 + 16];
        #pragma unroll
        for (int i = 0; i < 8; ++i) {
            const int r = rowBase + ((lane < 16) ? i : i + 8);
            float v0 = acc0[i] + bias0;
            float v1 = acc1[i] + bias1;
            out[(size_t)r * NHID + col0]      = v0 > 0.f ? v0 : 0.f;
            out[(size_t)r * NHID + col0 + 16] = v1 > 0.f ? v1 : 0.f;
        }
    }
}

extern "C" void kernel_launch(void* const* d_in, const int* in_sizes, int n_in,
                              void* d_out, int out_size, void* d_ws, size_t ws_size,
                              hipStream_t stream) {
    const float* A  = (const float*)d_in[0];  // appearance [N, 1024]
    const float* S  = (const float*)d_in[1];  // spatial    [N, 512]
    const float* W1 = (const float*)d_in[2];  // [16, 1024, 32]
    const float* b1 = (const float*)d_in[3];  // [16, 32]
    const float* W2 = (const float*)d_in[4];  // [16, 512, 32]
    const float* b2 = (const float*)d_in[5];  // [16, 32]
    const float* W3 = (const float*)d_in[6];  // [16, 32, 512]
    const float* b3 = (const float*)d_in[7];  // [16, 512]
    float* out = (float*)d_out;

    const int n = in_sizes[0] / NAPP;         // 65536
    const int blocks = n / M_TILE;            // 1024
    const size_t shmem = (size_t)(M_TILE * NCOLS + NCOLS + NHID) * sizeof(float); // 132 KB

    mbf_fused_kernel<<<blocks, BLOCK_THREADS, shmem, stream>>>(A, S, W1, b1, W2, b2, W3, b3, out);
}